// Entmax15_62551903699244
// MI455X (gfx1250) — compile-verified
//
#include <hip/hip_runtime.h>
#include <math.h>

#define NROWS 2048
#define VDIM  32000
#define CAP   4096
#define TPB   512

typedef __attribute__((ext_vector_type(4))) unsigned int v4u;
typedef __attribute__((ext_vector_type(8))) int          v8i;
typedef __attribute__((ext_vector_type(4))) int          v4i;
typedef __attribute__((ext_vector_type(4))) float        v4f;

__global__ __launch_bounds__(TPB)
void entmax15_kernel(const float* __restrict__ X, float* __restrict__ Y) {
    __shared__ float row[VDIM];      // 128000 B : whole row resident in LDS
    __shared__ float cand[CAP];      //  16384 B : shifted candidates Xs
    __shared__ float wred[TPB / 32];
    __shared__ int   s_cnt;
    __shared__ float s_max;
    __shared__ float s_tau;

    const int tid  = threadIdx.x;
    const int wid  = tid >> 5;
    const int lane = tid & 31;
    const float* xrow = X + (size_t)blockIdx.x * VDIM;
    float*       yrow = Y + (size_t)blockIdx.x * VDIM;

    // ---------------- Phase 0: DMA row -> LDS via Tensor Data Mover ----------
    if (tid == 0) {
        s_cnt = 0;
#if __has_builtin(__builtin_amdgcn_tensor_load_to_lds)
        unsigned long long ga = (unsigned long long)(uintptr_t)xrow;
        // flat shared pointer low 32 bits == LDS byte offset
        unsigned int ldsa = (unsigned int)(unsigned long long)(uintptr_t)(&row[0]);
        // D# group 0: [1:0]=count=1 | [63:32]=lds_addr | [120:64]=global_addr | [127:126]=type=2
        v4u g0 = { 1u,
                   ldsa,
                   (unsigned int)(ga & 0xFFFFFFFFull),
                   (unsigned int)((ga >> 32) & 0x1FFFFFFull) | (2u << 30) };
        // D# group 1: data_size=2 (4B); tensor_dim0=VDIM; tensor_dim1=1;
        //             tile_dim0=VDIM; tile_dim1=1; tensor_dim0_stride=VDIM
        v8i g1 = { (int)(2u << 16),
                   (int)((VDIM & 0xFFFF) << 16),
                   (int)((VDIM >> 16) | (1u << 16)),
                   (int)((VDIM & 0xFFFF) << 16),
                   1,
                   VDIM,
                   0,
                   0 };
        v4i gz = { 0, 0, 0, 0 };
#if __clang_major__ >= 23
        v8i g4 = { 0, 0, 0, 0, 0, 0, 0, 0 };
        __builtin_amdgcn_tensor_load_to_lds(g0, g1, gz, gz, g4, 0);
#else
        __builtin_amdgcn_tensor_load_to_lds(g0, g1, gz, gz, 0);
#endif
        __builtin_amdgcn_s_wait_tensorcnt(0);
#endif
    }
#if !__has_builtin(__builtin_amdgcn_tensor_load_to_lds)
    for (int i = tid * 4; i < VDIM; i += TPB * 4)
        *(v4f*)(&row[i]) = *(const v4f*)(&xrow[i]);
#endif
    __syncthreads();

    // ---------------- Phase 1: row max (LDS b128 reads, wave32 reduce) -------
    float m = -INFINITY;
    for (int i = tid * 4; i < VDIM; i += TPB * 4) {
        v4f v = *(const v4f*)(&row[i]);
        m = fmaxf(m, fmaxf(fmaxf(v.x, v.y), fmaxf(v.z, v.w)));
    }
    for (int off = 16; off > 0; off >>= 1) m = fmaxf(m, __shfl_xor(m, off, 32));
    if (lane == 0) wred[wid] = m;
    __syncthreads();
    if (tid < TPB / 32) {
        float t = wred[tid];
        for (int off = (TPB / 64); off > 0; off >>= 1) t = fmaxf(t, __shfl_xor(t, off, 32));
        if (tid == 0) s_max = t;
    }
    __syncthreads();

    // ---------------- Phase 2: candidates with X >= max-2 (tau* in [-1,0]) ---
    const float mx  = s_max;
    const float thr = mx - 2.0f;
    for (int i = tid * 4; i < VDIM; i += TPB * 4) {
        v4f v = *(const v4f*)(&row[i]);
        #pragma unroll
        for (int c = 0; c < 4; ++c) {
            float x = (c == 0) ? v.x : (c == 1) ? v.y : (c == 2) ? v.z : v.w;
            if (x >= thr) {
                int p = atomicAdd(&s_cnt, 1);       // ds_add_rtn_u32
                if (p < CAP) cand[p] = (x - mx) * 0.5f;
            }
        }
    }
    __syncthreads();

    const int n = min(s_cnt, CAP);
    int N = 2;
    while (N < n) N <<= 1;
    for (int i = n + tid; i < N; i += TPB) cand[i] = -INFINITY;
    __syncthreads();

    // ---------------- Phase 3: bitonic sort (descending) over N <= 4096 ------
    // Pair map: i = 2*idx - (idx & (j-1)); a wave (32 contiguous idx) touches an
    // aligned 64-element block when j <= 32 -> wave-local stage, no s_barrier
    // needed (wave32 lockstep + in-order LDS per wave). Full barrier only when
    // j >= 64 (cross-wave) or immediately after a cross-wave stage.
    bool prev_cross = false;
    for (int k = 2; k <= N; k <<= 1) {
        for (int j = k >> 1; j > 0; j >>= 1) {
            const bool cross = (j >= 64);
            if (cross || prev_cross) __syncthreads();
            else                     __builtin_amdgcn_wave_barrier();
            for (int idx = tid; idx < (N >> 1); idx += TPB) {
                int i   = 2 * idx - (idx & (j - 1));
                int ixj = i + j;
                float a = cand[i], b = cand[ixj];
                bool up  = ((i & k) == 0);              // descending blocks
                bool swp = up ? (a < b) : (a > b);
                if (swp) { cand[i] = b; cand[ixj] = a; }
            }
            prev_cross = cross;
        }
    }
    __syncthreads();

    // ---------------- Phase 4: exact reference recurrence (thread 0) ---------
    if (tid == 0) {
        int   C  = 0;
        float cs = 0.0f, css = 0.0f;
        for (int k = 1; k <= n; ++k) {
            float x = cand[k - 1];
            cs  += x;
            css += x * x;
            float rk    = (float)k;
            float mean  = cs / rk;
            float msq   = css / rk;
            float ss    = rk * (msq - mean * mean);
            float delta = (1.0f - ss) / rk;
            if (delta < 0.0f) delta = 0.0f;
            float tau = mean - sqrtf(delta);
            if (tau <= x) C++;
        }
        cs = 0.0f; css = 0.0f;
        for (int k = 1; k <= C; ++k) {
            float x = cand[k - 1];
            cs += x; css += x * x;
        }
        float rk    = (float)C;
        float mean  = cs / rk;
        float msq   = css / rk;
        float ss    = rk * (msq - mean * mean);
        float delta = (1.0f - ss) / rk;
        if (delta < 0.0f) delta = 0.0f;
        s_tau = mean - sqrtf(delta);
    }
    __syncthreads();

    // ------------- Phase 5: y = clip(Xs - tau, 0)^2, NT b128 stores ----------
    const float tau = s_tau;
    for (int i = tid * 4; i < VDIM; i += TPB * 4) {
        v4f v = *(const v4f*)(&row[i]);
        v4f o;
        o.x = fmaxf((v.x - mx) * 0.5f - tau, 0.0f); o.x *= o.x;
        o.y = fmaxf((v.y - mx) * 0.5f - tau, 0.0f); o.y *= o.y;
        o.z = fmaxf((v.z - mx) * 0.5f - tau, 0.0f); o.z *= o.z;
        o.w = fmaxf((v.w - mx) * 0.5f - tau, 0.0f); o.w *= o.w;
        __builtin_nontemporal_store(o, (v4f*)(&yrow[i]));   // TH_NT stream-out
    }
}

extern "C" void kernel_launch(void* const* d_in, const int* in_sizes, int n_in,
                              void* d_out, int out_size, void* d_ws, size_t ws_size,
                              hipStream_t stream) {
    (void)in_sizes; (void)n_in; (void)out_size; (void)d_ws; (void)ws_size;
    const float* X = (const float*)d_in[0];
    float*       Y = (float*)d_out;
    entmax15_kernel<<<dim3(NROWS), dim3(TPB), 0, stream>>>(X, Y);
}